// CenterLoss_35107062677983
// MI455X (gfx1250) — compile-verified
//
#include <hip/hip_runtime.h>

typedef float v2f __attribute__((ext_vector_type(2)));
typedef float v8f __attribute__((ext_vector_type(8)));

#define BATCH   2048
#define NCLASS  100000
#define DIM     64
#define TILES   (BATCH / 16)        // 128 tiles of 16 samples
#define WAVES_PER_BLOCK 8           // 256 threads = 8 wave32

__global__ void cl_zero_ws(float* ws) {
    if (threadIdx.x == 0 && blockIdx.x == 0) ws[0] = 0.0f;
}

// One wave per 16-sample tile.
// dist_i = || x_i - centers[label_i] ||^2 computed as diag( dX * dX^T )
// via V_WMMA_F32_16X16X4_F32, accumulating K = 0..63 in 16 steps of 4.
__global__ __launch_bounds__(256)
void cl_tile_kernel(const float* __restrict__ x,
                    const float* __restrict__ centers,
                    const int*   __restrict__ labels,
                    float*       __restrict__ accum) {
    const int lane = threadIdx.x & 31;
    const int tile = blockIdx.x * WAVES_PER_BLOCK + (threadIdx.x >> 5);
    const int row  = tile * 16 + (lane & 15);   // matrix row this lane feeds
    const int h    = lane >> 4;                 // half-select: K pair {0,1} or {2,3}

    const int   lbl = labels[row];
    const float* xr = x       + (size_t)row * DIM;
    const float* cr = centers + (size_t)lbl * DIM;

    // Build A fragments for dX = X - G.  Because B = A^T for dX*dX^T, the
    // ISA's 16x4 A-layout fragment is identical to the 4x16 B-layout
    // fragment, so one register pair serves both WMMA operands.
    v2f fr[16];
#pragma unroll
    for (int k = 0; k < 16; ++k) {
        const int off = 4 * k + 2 * h;
        v2f xv = *(const v2f*)(xr + off);
        v2f cv = *(const v2f*)(cr + off);
        fr[k].x = xv.x - cv.x;
        fr[k].y = xv.y - cv.y;
    }

    v8f acc = {0.f, 0.f, 0.f, 0.f, 0.f, 0.f, 0.f, 0.f};
#pragma unroll
    for (int k = 0; k < 16; ++k) {
        // 8 args: (neg_a, A, neg_b, B, c_mod, C, reuse_a, reuse_b)
        acc = __builtin_amdgcn_wmma_f32_16x16x4_f32(
            false, fr[k], false, fr[k], (short)0, acc, false, false);
    }

    // Diagonal of the 16x16 f32 D matrix:
    //   i in [0,8):  lane = i,      value = acc[i]
    //   i in [8,16): lane = i + 16, value = acc[lane - 24]
    const bool active = (lane < 8) || (lane >= 24);
    const int  sel    = (lane < 8) ? lane : (lane - 24);
    float v = acc[0];
#pragma unroll
    for (int j = 1; j < 8; ++j) v = (sel == j) ? acc[j] : v;

    float part = active ? fminf(fmaxf(v, 1e-12f), 1e12f) : 0.0f;

    // wave32 butterfly reduction
#pragma unroll
    for (int off = 16; off > 0; off >>= 1)
        part += __shfl_xor(part, off, 32);

    if (lane == 0) atomicAdd(accum, part);
}

__global__ void cl_finalize(const float* __restrict__ ws,
                            float* __restrict__ out) {
    if (threadIdx.x == 0 && blockIdx.x == 0) {
        // masked-out entries each contribute 1e-12 after the clip:
        // B*(C-1)*1e-12 / B = (C-1)*1e-12
        out[0] = ws[0] / (float)BATCH + (float)(NCLASS - 1) * 1e-12f;
    }
}

extern "C" void kernel_launch(void* const* d_in, const int* in_sizes, int n_in,
                              void* d_out, int out_size, void* d_ws, size_t ws_size,
                              hipStream_t stream) {
    const float* x       = (const float*)d_in[0];
    const float* centers = (const float*)d_in[1];
    const int*   labels  = (const int*)d_in[2];
    float*       out     = (float*)d_out;
    float*       ws      = (float*)d_ws;

    cl_zero_ws<<<1, 32, 0, stream>>>(ws);
    cl_tile_kernel<<<TILES / WAVES_PER_BLOCK, 32 * WAVES_PER_BLOCK, 0, stream>>>(
        x, centers, labels, ws);
    cl_finalize<<<1, 32, 0, stream>>>(ws, out);
}